// EncoderALSTM_72799695667471
// MI455X (gfx1250) — compile-verified
//
#include <hip/hip_runtime.h>
#include <hip/hip_bf16.h>
#include <cstddef>

// Sizes fixed by the reference
#define T_  128
#define B_  32
#define H_  256
#define I_  256
#define G4H 1024      // 4*H

typedef __attribute__((__ext_vector_type__(2))) float v2f;
typedef __attribute__((__ext_vector_type__(8))) float v8f;

__device__ __forceinline__ float sigm(float x) { return 1.0f / (1.0f + __expf(-x)); }

// -----------------------------------------------------------------------------
// Kernel 1: parallel precompute of the input-gate contribution for all steps.
//   G[row, n] = emb[row, :] @ W_ih^T[:, n] + b_ih[n] + b_hh[n],  row = t*B+b
// 4096x1024x256 GEMM in fp32 WMMA (16x16x4). 8 waves/block, 1 tile/wave.
// -----------------------------------------------------------------------------
__global__ __launch_bounds__(256) void gate_precompute(
    const float* __restrict__ emb,   // [4096][256]
    const float* __restrict__ W_ih,  // [1024][256]
    const float* __restrict__ b_ih,  // [1024]
    const float* __restrict__ b_hh,  // [1024]
    float* __restrict__ G)           // [4096][1024]
{
  const int lane  = threadIdx.x & 31;
  const int wave  = threadIdx.x >> 5;
  const int tile  = blockIdx.x * 8 + wave;   // 16384 tiles
  const int mtile = tile >> 6;               // 0..255
  const int ntile = tile & 63;               // 0..63
  const int lm    = lane & 15;
  const int koff  = (lane >> 4) * 2;

  const float* arow = emb  + (size_t)(mtile * 16 + lm) * I_;
  const float* brow = W_ih + (size_t)(ntile * 16 + lm) * I_;   // B[k][n] = W_ih[n][k]

  v8f acc = {};
  for (int k0 = 0; k0 < I_; k0 += 4) {
    v2f a = *(const v2f*)(arow + k0 + koff);
    v2f b = *(const v2f*)(brow + k0 + koff);
    acc = __builtin_amdgcn_wmma_f32_16x16x4_f32(false, a, false, b,
                                                (short)0, acc, false, false);
  }
  const int n     = ntile * 16 + lm;
  const float bia = b_ih[n] + b_hh[n];
  const int mhi   = (lane >> 4) * 8;
  for (int r = 0; r < 8; ++r) {
    int m = mtile * 16 + r + mhi;
    G[(size_t)m * G4H + n] = acc[r] + bia;
  }
}

// -----------------------------------------------------------------------------
// Kernel 2: persistent single-workgroup recurrence (1024 thr = 32 wave32).
// Per step t: attention over buf[0..t-1] (buf == out), Luong combine via W_a
// (fp32 WMMA), gates = G[t] + h @ W_hh^T (fp32 WMMA, done in two 512-col
// halves {i,g} then {f,o} sharing one 64KB LDS buffer), LSTM elementwise.
// -----------------------------------------------------------------------------
__global__ __launch_bounds__(1024) void alstm_recurrent(
    const float* __restrict__ G,     // [4096][1024]
    const float* __restrict__ W_hh,  // [1024][256]
    const float* __restrict__ W_a,   // [256][512]
    const float* __restrict__ b_a,   // [256]
    const int*   __restrict__ lens,  // [32]
    float* __restrict__ out)         // [T*B*H] outputs ++ [B*H] h_fin ++ [B*H] c_fin
{
  __shared__ float h_sh[B_][H_];       // 32 KB: current hidden state
  __shared__ float sc_sh[T_][B_];      // 16 KB: scores -> softmax alpha
  __shared__ float buf2[B_][2 * H_];   // 64 KB: [ctx||h] A-matrix / gate halves

  const int tid  = threadIdx.x;
  const int lane = tid & 31;
  const int wave = tid >> 5;
  const int lm   = lane & 15;
  const int koff = (lane >> 4) * 2;
  const int mhi  = (lane >> 4) * 8;
  // Elementwise ownership: thread owns (b0 + 4j, hh) for j=0..7
  const int hh = tid & 255;
  const int b0 = tid >> 8;

  float c_reg[8], ct_reg[8], hf_reg[8], cf_reg[8], h_new[8];
  for (int j = 0; j < 8; ++j) { c_reg[j] = ct_reg[j] = hf_reg[j] = cf_reg[j] = 0.f; }
  for (int e = tid; e < B_ * H_; e += 1024) ((float*)h_sh)[e] = 0.f;
  __syncthreads();

  for (int t = 0; t < T_; ++t) {
    // ---------------- Phase A: attention + combine (only t > 0) --------------
    if (t > 0) {
      // A1: scores[t'][b] = <buf[t'][b], h[b]>
      for (int idx = tid; idx < t * B_; idx += 1024) {
        int tp = idx >> 5, bb = idx & 31;
        const float* brow = out + ((size_t)tp * B_ + bb) * H_;
        const float* hrow = h_sh[bb];
        float s = 0.f;
        for (int k = 0; k < H_; k += 4) {
          float4 x = *(const float4*)(brow + k);
          s += x.x * hrow[k] + x.y * hrow[k + 1] + x.z * hrow[k + 2] + x.w * hrow[k + 3];
        }
        sc_sh[tp][bb] = s;
      }
      __syncthreads();
      // A2: softmax along t' (one thread per batch element)
      if (tid < B_) {
        float mx = -3.0e38f;
        for (int tp = 0; tp < t; ++tp) mx = fmaxf(mx, sc_sh[tp][tid]);
        float sum = 0.f;
        for (int tp = 0; tp < t; ++tp) {
          float e = __expf(sc_sh[tp][tid] - mx);
          sc_sh[tp][tid] = e;
          sum += e;
        }
        float inv = 1.f / sum;
        for (int tp = 0; tp < t; ++tp) sc_sh[tp][tid] *= inv;
      }
      __syncthreads();
      // A3: ctx = sum_t' alpha * buf[t'];  buf2 = [ctx || h]
      for (int e = tid; e < B_ * H_; e += 1024) {
        int bb = e >> 8, hc = e & 255;
        float acc = 0.f;
        for (int tp = 0; tp < t; ++tp)
          acc += sc_sh[tp][bb] * out[((size_t)tp * B_ + bb) * H_ + hc];
        buf2[bb][hc]      = acc;
        buf2[bb][H_ + hc] = h_sh[bb][hc];
      }
      __syncthreads();
      // A4: h = tanh(buf2[32,512] @ W_a^T + b_a). 32 tiles, one per wave.
      {
        const int mtile = wave & 1, ntile = wave >> 1;
        const float* arow = &buf2[mtile * 16 + lm][0];
        const float* brow = W_a + (size_t)(ntile * 16 + lm) * (2 * H_);
        v8f acc = {};
        for (int k0 = 0; k0 < 2 * H_; k0 += 4) {
          v2f a = *(const v2f*)(arow + k0 + koff);
          v2f b = *(const v2f*)(brow + k0 + koff);
          acc = __builtin_amdgcn_wmma_f32_16x16x4_f32(false, a, false, b,
                                                      (short)0, acc, false, false);
        }
        const int nc = ntile * 16 + lm;
        const float ba = b_a[nc];
        for (int r = 0; r < 8; ++r)
          h_sh[mtile * 16 + r + mhi][nc] = tanhf(acc[r] + ba);
      }
      __syncthreads();
    }

    // ---------------- Phase B: gates = G[t] + h @ W_hh^T ---------------------
    const float* Gt = G + (size_t)t * B_ * G4H;
    for (int half = 0; half < 2; ++half) {
      // 64 tiles per half (M=32, N=512 local); each wave does 2 tiles.
      for (int tt = wave; tt < 64; tt += 32) {
        const int mtile = tt & 1;
        const int nloc  = (tt >> 1) * 16 + lm;                         // 0..511
        const int gcol  = half * H_ + ((nloc < H_) ? nloc : H_ + nloc); // i/f then g/o
        const float* arow = h_sh[mtile * 16 + lm];
        const float* brow = W_hh + (size_t)gcol * H_;                  // B[k][n]=W_hh[n][k]
        v8f acc = {};
        for (int k0 = 0; k0 < H_; k0 += 4) {
          v2f a = *(const v2f*)(arow + k0 + koff);
          v2f b = *(const v2f*)(brow + k0 + koff);
          acc = __builtin_amdgcn_wmma_f32_16x16x4_f32(false, a, false, b,
                                                      (short)0, acc, false, false);
        }
        for (int r = 0; r < 8; ++r) {
          int m = mtile * 16 + r + mhi;
          buf2[m][nloc] = acc[r] + Gt[(size_t)m * G4H + gcol];
        }
      }
      __syncthreads();
      if (half == 0) {                       // consume i, g
        for (int j = 0; j < 8; ++j) {
          int bb = b0 + j * 4;
          ct_reg[j] = sigm(buf2[bb][hh]) * tanhf(buf2[bb][H_ + hh]);
        }
        __syncthreads();                     // readers done before f/o overwrite
      } else {                               // consume f, o -> c, h
        for (int j = 0; j < 8; ++j) {
          int bb = b0 + j * 4;
          float cn = sigm(buf2[bb][hh]) * c_reg[j] + ct_reg[j];
          c_reg[j] = cn;
          h_new[j] = sigm(buf2[bb][H_ + hh]) * tanhf(cn);
        }
        for (int j = 0; j < 8; ++j) {
          int bb = b0 + j * 4;
          float hn = h_new[j];
          h_sh[bb][hh] = hn;                               // next-step state
          out[((size_t)t * B_ + bb) * H_ + hh] = hn;       // buf[t] / output
          if (t == lens[bb] - 1) { hf_reg[j] = hn; cf_reg[j] = c_reg[j]; }
        }
      }
    }
    __threadfence();     // publish buf[t] (global) to the whole workgroup
    __syncthreads();
  }

  // Final states
  const size_t fin = (size_t)T_ * B_ * H_;
  for (int j = 0; j < 8; ++j) {
    int bb = b0 + j * 4;
    out[fin + (size_t)bb * H_ + hh]                      = hf_reg[j];
    out[fin + (size_t)B_ * H_ + (size_t)bb * H_ + hh]    = cf_reg[j];
  }
}

// -----------------------------------------------------------------------------
extern "C" void kernel_launch(void* const* d_in, const int* in_sizes, int n_in,
                              void* d_out, int out_size, void* d_ws, size_t ws_size,
                              hipStream_t stream) {
  const float* embs = (const float*)d_in[0];   // [128,32,256]
  const int*   lens = (const int*)  d_in[1];   // [32]
  const float* W_ih = (const float*)d_in[2];   // [1024,256]
  const float* W_hh = (const float*)d_in[3];   // [1024,256]
  const float* b_ih = (const float*)d_in[4];   // [1024]
  const float* b_hh = (const float*)d_in[5];   // [1024]
  const float* W_a  = (const float*)d_in[6];   // [256,512]
  const float* b_a  = (const float*)d_in[7];   // [256]
  float* out = (float*)d_out;
  float* G   = (float*)d_ws;                   // 4096*1024 f32 = 16 MB scratch

  gate_precompute<<<2048, 256, 0, stream>>>(embs, W_ih, b_ih, b_hh, G);
  alstm_recurrent<<<1, 1024, 0, stream>>>(G, W_hh, W_a, b_a, lens, out);
}